// IngredientsLoss_26465588478300
// MI455X (gfx1250) — compile-verified
//
#include <hip/hip_runtime.h>

#define N_FOODS     223
#define N_FOODS_PAD 224   // padded so the inner loop unrolls evenly (4x56)
#define N_CATS      5
#define ELEMS       210   // 7*3*10 elements per batch
#define BATCH       1024
#define BLOCK       224   // 7 waves (wave32): only 14 idle lanes vs 46 at 256
#define NWAVES      (BLOCK / 32)

typedef float v2f __attribute__((ext_vector_type(2)));
typedef float v8f __attribute__((ext_vector_type(8)));

// Full 32-lane sum using V_WMMA_F32_16X16X4_F32.
// A layout (ISA 7.12.2, 32-bit A 16x4): lane m (0-15) VGPR0 -> A[m][0],
// lane m+16 VGPR0 -> A[m][2]; we zero VGPR1 so A[m][1]=A[m][3]=0.
// B = all-ones (identical value in every slot => B's lane layout is irrelevant).
// D[m][n] = A[m][0]+A[m][2] = x[m]+x[m+16], replicated over n.
// D layout: VGPR v holds s[v] in lanes 0-15 and s[v+8] in lanes 16-31, so
// summing my 8 D components gives half the total; an xor-16 shuffle finishes.
__device__ __forceinline__ float wave_sum_wmma(float x) {
  v2f a; a.x = x;    a.y = 0.0f;
  v2f b; b.x = 1.0f; b.y = 1.0f;
  v8f c = {};
  v8f d = __builtin_amdgcn_wmma_f32_16x16x4_f32(
      /*neg_a=*/false, a, /*neg_b=*/false, b,
      /*c_mod=*/(short)0, c, /*reuse_a=*/false, /*reuse_b=*/false);
  float h = ((d[0] + d[1]) + (d[2] + d[3])) + ((d[4] + d[5]) + (d[6] + d[7]));
  h += __shfl_xor(h, 16, 32);
  return h;
}

__global__ void zero_out_kernel(float* out) {
  if (threadIdx.x == 0 && blockIdx.x == 0) out[0] = 0.0f;
}

__global__ void __launch_bounds__(BLOCK)
ingredients_loss_kernel(const float* __restrict__ y_pred,
                        const float* __restrict__ y,
                        const float* __restrict__ data,
                        float* __restrict__ out) {
  __shared__ float2 vc[N_CATS][N_FOODS_PAD]; // (val, center=i*val), 9.0 KB
  __shared__ float  wsum[NWAVES][N_CATS];
  __shared__ float  cabs[N_CATS];

  const int tid = threadIdx.x;
  const int b   = blockIdx.x;

  // Cooperative table build: data is row-major (food, cat). Pad entry is
  // (0,0): contributes exactly 0*(...) == 0 to every accumulation.
  for (int idx = tid; idx < N_FOODS_PAD * N_CATS; idx += BLOCK) {
    const int food = idx / N_CATS;
    const int cat  = idx % N_CATS;
    float2 e;
    if (food < N_FOODS) {
      const float v = data[idx];
      e = make_float2(v, (float)food * v);
    } else {
      e = make_float2(0.0f, 0.0f);
    }
    vc[cat][food] = e;
  }
  __syncthreads();

  float acc[N_CATS] = {0.f, 0.f, 0.f, 0.f, 0.f};

  if (tid < ELEMS) {
    const int flat = (b * ELEMS + tid) * 2;  // F=2, take [...,0]
    const float gx = y[flat];
    // round-half-even (matches jnp.round), then soft bound
    const float r  = rintf(y_pred[flat]);
    // sigmoid(50*(222.5-r)) = 1/(1+exp2(-50*(222.5-r)*log2e))
    const float s  = 50.0f * (222.5f - r) * 1.44269504088896f;
    const float px = r / (1.0f + __builtin_amdgcn_exp2f(-s));

    const float K = -144.269504088896f;  // -(1/SIGMA2)/ln2 = -100*log2(e)

    #pragma unroll 1
    for (int cat = 0; cat < N_CATS; ++cat) {
      float a = 0.0f;
      #pragma unroll 4
      for (int i = 0; i < N_FOODS_PAD; ++i) {   // 224 = 4*56, no remainder
        const float2 f = vc[cat][i];            // broadcast LDS b64 load
        const float dp = px - f.y;
        const float dg = gx - f.y;
        const float ep = __builtin_amdgcn_exp2f(K * dp * dp);  // v_exp_f32
        const float eg = __builtin_amdgcn_exp2f(K * dg * dg);  // v_exp_f32
        a = fmaf(f.x, ep - eg, a);              // fused pred-gold diff
      }
      acc[cat] = a;
    }
  }
  // Uniform control flow here: EXEC all ones, as WMMA requires.
  const int wave = tid >> 5;
  const int lane = tid & 31;
  #pragma unroll
  for (int cat = 0; cat < N_CATS; ++cat) {
    const float t = wave_sum_wmma(acc[cat]);
    if (lane == 0) wsum[wave][cat] = t;
  }
  __syncthreads();

  if (tid < N_CATS) {
    float s = 0.0f;
    #pragma unroll
    for (int w = 0; w < NWAVES; ++w) s += wsum[w][tid];
    cabs[tid] = fabsf(s);   // |pred_sum - gold_sum| for this (batch, cat)
  }
  __syncthreads();

  if (tid == 0) {
    const float tot = ((cabs[0] + cabs[1]) + (cabs[2] + cabs[3])) + cabs[4];
    atomicAdd(out, tot * (3.0f / 1024.0f));  // PENALTY / B, one atomic/block
  }
}

extern "C" void kernel_launch(void* const* d_in, const int* in_sizes, int n_in,
                              void* d_out, int out_size, void* d_ws, size_t ws_size,
                              hipStream_t stream) {
  (void)in_sizes; (void)n_in; (void)out_size; (void)d_ws; (void)ws_size;
  const float* y_pred = (const float*)d_in[0];
  const float* yy     = (const float*)d_in[1];
  const float* data   = (const float*)d_in[2];
  float* out          = (float*)d_out;

  zero_out_kernel<<<1, 32, 0, stream>>>(out);
  ingredients_loss_kernel<<<BATCH, BLOCK, 0, stream>>>(y_pred, yy, data, out);
}